// HybridQuantumClassifier_65481071396625
// MI455X (gfx1250) — compile-verified
//
#include <hip/hip_runtime.h>
#include <math.h>
#include <stdint.h>

typedef float v2f __attribute__((ext_vector_type(2)));
typedef float v8f __attribute__((ext_vector_type(8)));

#define NPIX   784      // 28*28 pixels per image
#define KCHUNK 112      // K tile staged in LDS (4 pixel-rows of 28)
#define PITCH  116      // LDS row pitch (floats): 16B aligned, conflict-free b64 reads
#define WAVES  8        // waves per block (wave32)
#define IMGS   16       // images per wave (WMMA M)

__global__ __launch_bounds__(256)
void hqc_kernel(const float* __restrict__ x, const float* __restrict__ W,
                float* __restrict__ out, int nImages)
{
    __shared__ float tiles[WAVES][IMGS * PITCH];   // per-wave A staging (~58 KB total)

    const int tid  = threadIdx.x;
    const int lane = tid & 31;
    const int wave = tid >> 5;

    // Per-parity conv weights, 1/784 folded in. W flat: [o][kh][kw] = o*4+kh*2+kw.
    const float inv = 1.0f / (float)NPIX;
    const float S00 = W[0] + W[4] + W[8]  + W[12];
    const float S01 = W[1] + W[5] + W[9]  + W[13];
    const float S10 = W[2] + W[6] + W[10] + W[14];
    const float S11 = W[3] + W[7] + W[11] + W[15];
    // B fragment constants: pixel-row-even -> (S00,S01), row-odd -> (S10,S11).
    const v2f bEven = { S00 * inv, S01 * inv };
    const v2f bOdd  = { S10 * inv, S11 * inv };

    const int tile = blockIdx.x * WAVES + wave;        // 16-image tile id
    const long long imgBase = (long long)tile * IMGS;
    if (imgBase >= nImages) return;                    // uniform per wave; no barriers used

    float* smem = tiles[wave];
    // Low 32 bits of the flat pointer == LDS byte offset (flat->LDS aperture rule).
    const unsigned ldsBase = (unsigned)(uint64_t)(uintptr_t)smem;
    const int g = lane >> 4;                           // lane group -> K pair
    const int m = lane & 15;                           // A row / D column index

    v8f acc = {};

    #pragma unroll 1
    for (int chunk = 0; chunk < NPIX / KCHUNK; ++chunk) {
        const int cb = chunk * KCHUNK;

        // WAR guard: previous chunk's ds_load reads must retire before the
        // async engine overwrites the LDS tile (async LDS writes are unordered
        // with DS-pipe ops).
        asm volatile("s_wait_dscnt 0x0" ::: "memory");

        // ---- stage 16 images x 112 floats: 16 async B128 copies, no waits
        // between them -> 16 outstanding HBM requests per wave ----
        if (lane < 28) {                               // 28 float4 per image row-segment
            uint64_t gp = (uint64_t)(uintptr_t)(x + imgBase * NPIX + cb + lane * 4);
            unsigned lp = ldsBase + (unsigned)(lane * 16);
            #pragma unroll
            for (int img = 0; img < IMGS; ++img) {
                asm volatile("global_load_async_to_lds_b128 %0, %1, off"
                             :: "v"(lp), "v"(gp) : "memory");
                gp += (uint64_t)NPIX * 4;              // next image, same segment
                lp += PITCH * 4;                       // next LDS row
            }
        }
        asm volatile("s_wait_asynccnt 0x0" ::: "memory");

        // ---- 28 WMMA steps of K=4; B comes from registers (2 constants) ----
        #pragma unroll
        for (int s = 0; s < 28; ++s) {
            const int k0 = s * 4 + 2 * g;              // this lane's K pair
            v2f a = *(const v2f*)(smem + m * PITCH + k0);       // ds_load_b64
            v2f b = (((s * 4) / 28) & 1) ? bOdd : bEven;        // compile-time select
            acc = __builtin_amdgcn_wmma_f32_16x16x4_f32(
                      false, a, false, b, (short)0, acc, false, false);
        }
    }

    // D layout: lane L, VGPR v -> M = 8*(L>>4)+v, N = L&15; all N columns equal.
    // Lanes 0 and 16 publish their 8 images each. |angle| << 1, so the
    // hardware transcendentals (v_sin_f32 / v_exp_f32) are plenty accurate.
    if (m == 0) {
        #pragma unroll
        for (int v = 0; v < 8; ++v) {
            long long img = imgBase + 8 * g + v;
            float s  = __sinf(acc[v]);                 // z = -sin(theta)
            float e  = __expf(s);                      // exp(-z) = exp(sin)
            float p  = 1.0f / (1.0f + e);              // sigmoid(-sin)
            float2 r = make_float2(p, 1.0f - p);
            *(float2*)(out + img * 2) = r;             // global_store_b64
        }
    }
}

extern "C" void kernel_launch(void* const* d_in, const int* in_sizes, int n_in,
                              void* d_out, int out_size, void* d_ws, size_t ws_size,
                              hipStream_t stream)
{
    const float* x = (const float*)d_in[0];   // (65536,1,28,28) f32
    const float* W = (const float*)d_in[1];   // (4,1,2,2) f32
    float* out = (float*)d_out;               // (65536,2) f32

    const int nImages = in_sizes[0] / NPIX;                      // 65536
    const int imgsPerBlock = WAVES * IMGS;                       // 128
    const int blocks = (nImages + imgsPerBlock - 1) / imgsPerBlock;  // 512

    hqc_kernel<<<blocks, 256, 0, stream>>>(x, W, out, nImages);
}